// Attention_76020921139645
// MI455X (gfx1250) — compile-verified
//
#include <hip/hip_runtime.h>
#include <math.h>

#define B_   4
#define T_   128
#define S_   512
#define D_   512
#define BT_  (B_ * T_)
#define BS_  (B_ * S_)

typedef __attribute__((ext_vector_type(2))) float v2f;
typedef __attribute__((ext_vector_type(8))) float v8f;

__device__ __forceinline__ v8f wmma4(v2f a, v2f b, v8f c) {
    // D = A(16x4) * B(4x16) + C, all f32
    return __builtin_amdgcn_wmma_f32_16x16x4_f32(
        /*neg_a=*/false, a, /*neg_b=*/false, b,
        /*c_mod=*/(short)0, c, /*reuse_a=*/false, /*reuse_b=*/false);
}

// gfx1250 hardware tanh (TRANS pipe, co-executes with VALU); guarded fallback.
__device__ __forceinline__ float fast_tanh(float x) {
#if __has_builtin(__builtin_amdgcn_tanhf)
    return __builtin_amdgcn_tanhf(x);
#else
    return tanhf(x);
#endif
}

// -------------------------------------------------------------------------
// GEMM: out[m][n] = sum_k A[m*K+k] * W[n*K+k] (+ bias[n])
// A: (M,K) row-major, W: (N,K) row-major (both K-contiguous).
// One wave -> one 16x32 strip (two 16x16 tiles sharing the A fragment).
// -------------------------------------------------------------------------
__global__ void gemm_nk2_kernel(const float* __restrict__ A,
                                const float* __restrict__ W,
                                const float* __restrict__ bias,
                                float* __restrict__ out,
                                int M, int N, int K) {
    const int wave = threadIdx.x >> 5;
    const int lane = threadIdx.x & 31;
    const int pairsN = N >> 5;                    // pairs of 16-wide tiles
    const int job = blockIdx.x * (blockDim.x >> 5) + wave;
    if (job >= (M >> 4) * pairsN) return;         // wave-uniform
    const int tm = job / pairsN;
    const int tp = job - tm * pairsN;             // N-pair index
    const int half = lane >> 4;
    const int r    = lane & 15;

    const float* Arow  = A + (size_t)(tm * 16 + r) * K + 2 * half;
    const float* Wrow0 = W + (size_t)(tp * 32 + r) * K + 2 * half;
    const float* Wrow1 = Wrow0 + (size_t)16 * K;

    v8f acc0 = {};
    v8f acc1 = {};
#pragma unroll 4
    for (int k = 0; k < K; k += 4) {
        v2f a  = *(const v2f*)(Arow + k);
        v2f b0 = *(const v2f*)(Wrow0 + k);
        v2f b1 = *(const v2f*)(Wrow1 + k);
        acc0 = wmma4(a, b0, acc0);
        acc1 = wmma4(a, b1, acc1);
    }

    const float bv0 = bias ? bias[tp * 32 + r]      : 0.0f;
    const float bv1 = bias ? bias[tp * 32 + 16 + r] : 0.0f;
    float* orow0 = out + (size_t)(tm * 16 + 8 * half) * N + tp * 32 + r;
    float* orow1 = orow0 + 16;
#pragma unroll
    for (int j = 0; j < 8; ++j) {
        orow0[(size_t)j * N] = acc0[j] + bv0;
        orow1[(size_t)j * N] = acc1[j] + bv1;
    }
}

// -------------------------------------------------------------------------
// Fused additive-attention score + softmax.
// One block per (b,t): align[b,t,s] = softmax_s( sum_d v[d]*tanh(wq+uh) )
// -------------------------------------------------------------------------
__global__ void score_softmax_kernel(const float* __restrict__ wq,
                                     const float* __restrict__ uh,
                                     const float* __restrict__ v,
                                     float* __restrict__ align) {
    const int bt = blockIdx.x;          // 0..BT-1
    const int b  = bt / T_;
    const int tid = threadIdx.x;        // 256 threads = 8 waves
    const int wave = tid >> 5;
    const int lane = tid & 31;

    __shared__ float s_wq[D_];
    __shared__ float s_v[D_];
    __shared__ float s_sc[S_];
    __shared__ float red[256];

    for (int d = tid; d < D_; d += 256) {
        s_wq[d] = wq[(size_t)bt * D_ + d];
        s_v[d]  = v[d];
    }
    __syncthreads();

    // each wave owns s = wave, wave+8, ... ; lanes split the D reduction
    for (int s = wave; s < S_; s += 8) {
        const float* uhrow = uh + ((size_t)b * S_ + s) * D_;
        float acc = 0.0f;
#pragma unroll 4
        for (int d = lane; d < D_; d += 32)
            acc += s_v[d] * fast_tanh(s_wq[d] + uhrow[d]);
        for (int off = 16; off > 0; off >>= 1)
            acc += __shfl_xor(acc, off, 32);
        if (lane == 0) s_sc[s] = acc;
    }
    __syncthreads();

    // softmax over s_sc[0..S-1]
    float m = -INFINITY;
    for (int s = tid; s < S_; s += 256) m = fmaxf(m, s_sc[s]);
    red[tid] = m;
    __syncthreads();
    for (int off = 128; off > 0; off >>= 1) {
        if (tid < off) red[tid] = fmaxf(red[tid], red[tid + off]);
        __syncthreads();
    }
    const float maxv = red[0];
    __syncthreads();

    float sum = 0.0f;
    for (int s = tid; s < S_; s += 256) {
        const float e = __expf(s_sc[s] - maxv);
        s_sc[s] = e;
        sum += e;
    }
    red[tid] = sum;
    __syncthreads();
    for (int off = 128; off > 0; off >>= 1) {
        if (tid < off) red[tid] += red[tid + off];
        __syncthreads();
    }
    const float inv = 1.0f / red[0];

    for (int s = tid; s < S_; s += 256)
        align[(size_t)bt * S_ + s] = s_sc[s] * inv;
}

// -------------------------------------------------------------------------
// Batched GEMM: c[b] (T x D) = P[b] (T x S) @ ctx[b] (S x D)
// B-matrix is (K,N) with N contiguous -> strided b32 fragment loads
// (coalesced across lanes since n = lane).
// -------------------------------------------------------------------------
__global__ void gemm_kn_batched_kernel(const float* __restrict__ P,
                                       const float* __restrict__ Ctx,
                                       float* __restrict__ out) {
    const int b = blockIdx.y;
    const float* A  = P   + (size_t)b * T_ * S_;
    const float* Bm = Ctx + (size_t)b * S_ * D_;
    float* O        = out + (size_t)b * T_ * D_;

    const int wave = threadIdx.x >> 5;
    const int lane = threadIdx.x & 31;
    const int tilesN = D_ >> 4;
    const int tile = blockIdx.x * (blockDim.x >> 5) + wave;
    if (tile >= (T_ >> 4) * tilesN) return;   // wave-uniform
    const int tm = tile / tilesN;
    const int tn = tile - tm * tilesN;
    const int half = lane >> 4;
    const int r    = lane & 15;

    const float* Arow = A  + (size_t)(tm * 16 + r) * S_ + 2 * half;
    const float* Bcol = Bm + (size_t)(2 * half) * D_ + tn * 16 + r;

    v8f acc = {};
#pragma unroll 4
    for (int k = 0; k < S_; k += 4) {
        v2f a = *(const v2f*)(Arow + k);
        v2f bb;
        bb.x = Bcol[(size_t)k * D_];
        bb.y = Bcol[(size_t)(k + 1) * D_];
        acc = wmma4(a, bb, acc);
    }

    float* orow = O + (size_t)(tm * 16 + 8 * half) * D_ + tn * 16 + r;
#pragma unroll
    for (int j = 0; j < 8; ++j)
        orow[(size_t)j * D_] = acc[j];
}

// -------------------------------------------------------------------------
// Output projection without materializing the concat:
// attn[m][n] = sum_k c[m][k]*Wout[n][k] + sum_k inp[m][k]*Wout[n][D+k] + bout[n]
// Wout: (D, 2D) row-major.
// -------------------------------------------------------------------------
__global__ void gemm_out_kernel(const float* __restrict__ Cv,
                                const float* __restrict__ Inp,
                                const float* __restrict__ Wout,
                                const float* __restrict__ bout,
                                float* __restrict__ out) {
    const int wave = threadIdx.x >> 5;
    const int lane = threadIdx.x & 31;
    const int tilesN = D_ >> 4;
    const int tile = blockIdx.x * (blockDim.x >> 5) + wave;
    if (tile >= (BT_ >> 4) * tilesN) return;   // wave-uniform
    const int tm = tile / tilesN;
    const int tn = tile - tm * tilesN;
    const int half = lane >> 4;
    const int r    = lane & 15;

    const float* A1   = Cv  + (size_t)(tm * 16 + r) * D_ + 2 * half;
    const float* A2   = Inp + (size_t)(tm * 16 + r) * D_ + 2 * half;
    const float* Wrow = Wout + (size_t)(tn * 16 + r) * (2 * D_) + 2 * half;

    v8f acc = {};
#pragma unroll 8
    for (int k = 0; k < D_; k += 4) {
        v2f a = *(const v2f*)(A1 + k);
        v2f b = *(const v2f*)(Wrow + k);
        acc = wmma4(a, b, acc);
    }
#pragma unroll 8
    for (int k = 0; k < D_; k += 4) {
        v2f a = *(const v2f*)(A2 + k);
        v2f b = *(const v2f*)(Wrow + D_ + k);
        acc = wmma4(a, b, acc);
    }

    const float bval = bout[tn * 16 + r];
    float* orow = out + (size_t)(tm * 16 + 8 * half) * D_ + tn * 16 + r;
#pragma unroll
    for (int j = 0; j < 8; ++j)
        orow[(size_t)j * D_] = acc[j] + bval;
}

// -------------------------------------------------------------------------
extern "C" void kernel_launch(void* const* d_in, const int* in_sizes, int n_in,
                              void* d_out, int out_size, void* d_ws, size_t ws_size,
                              hipStream_t stream) {
    const float* inp  = (const float*)d_in[0];   // (B,T,D)
    const float* ctx  = (const float*)d_in[1];   // (B,S,D)
    const float* Wq   = (const float*)d_in[2];   // (D,D)
    const float* bq   = (const float*)d_in[3];   // (D)
    const float* Wc   = (const float*)d_in[4];   // (D,D)
    const float* v    = (const float*)d_in[5];   // (D)
    const float* Wout = (const float*)d_in[6];   // (D,2D)
    const float* bout = (const float*)d_in[7];   // (D)

    float* attn  = (float*)d_out;                        // B*T*D
    float* align = (float*)d_out + (size_t)BT_ * D_;     // B*T*S

    float* wq = (float*)d_ws;                            // BT*D
    float* uh = wq + (size_t)BT_ * D_;                   // BS*D
    float* cv = uh + (size_t)BS_ * D_;                   // BT*D

    // 1) wq = inp @ Wq^T + bq     (M=BT, N=D, K=D): 32*16 = 512 strip-jobs
    gemm_nk2_kernel<<<64, 256, 0, stream>>>(inp, Wq, bq, wq, BT_, D_, D_);

    // 2) uh = ctx @ Wc^T          (M=BS, N=D, K=D): 128*16 = 2048 strip-jobs
    gemm_nk2_kernel<<<256, 256, 0, stream>>>(ctx, Wc, nullptr, uh, BS_, D_, D_);

    // 3) fused score + softmax -> align (written directly to d_out)
    score_softmax_kernel<<<BT_, 256, 0, stream>>>(wq, uh, v, align);

    // 4) c[b] = P[b] @ ctx[b]     (per batch: 256 tiles)
    gemm_kn_batched_kernel<<<dim3(32, B_), 256, 0, stream>>>(align, ctx, cv);

    // 5) attn = [c, inp] @ Wout^T + bout
    gemm_out_kernel<<<128, 256, 0, stream>>>(cv, inp, Wout, bout, attn);
}